// HypernetDecoder_28312424415688
// MI455X (gfx1250) — compile-verified
//
#include <hip/hip_runtime.h>

// ---------------------------------------------------------------------------
// HypernetDecoder for MI455X (gfx1250, wave32).
// Trunk MLP input z is broadcast over G in the reference, so trunk +
// hypernet-output GEMMs are computed for the 128 unique b-rows only
// (100x FLOP reduction). fp32 WMMA (v_wmma_f32_16x16x4_f32) keeps
// bit-faithful precision while using the CDNA5 matrix pipe.
// ---------------------------------------------------------------------------

typedef __attribute__((ext_vector_type(2))) float v2f;
typedef __attribute__((ext_vector_type(8))) float v8f;

#define LRELU_SLOPE 0.01f
#define OUT_PER_B   8450   // 64 + 1 + 4096 + 64 + 4096 + 64 + 64 + 1

// ---------------------------------------------------------------------------
// Wave-level WMMA f32 GEMM:  C = act(A @ B + bias)
//   A: MxK row-major, B: KxN row-major, bias: length N, C: MxN row-major
// One wave32 computes a 16x64 output tile (4 x v8f accumulators); the A
// fragment is loaded once per k-step and reused by 4 WMMAs.
// Edge tiles are handled by SHIFTING the tile origin (n0/m0 clamp) instead of
// masking: overlapping tiles recompute bit-identical values, so all loads and
// stores are unconditional and EXEC stays all-1s (required for WMMA).
// Requires N >= 64, M >= 16, K % 4 == 0.
// Fragment layouts per CDNA5 ISA 7.12.2 (f32):
//   A (16x4): lane l holds row M=l%16; VGPR v holds K = k0 + 2*(l/16) + v
//   B (4x16): lane l holds col N=l%16; VGPR v holds K = k0 + 2*(l/16) + v
//   C (16x16): VGPR v: lanes 0-15 -> M=v, lanes 16-31 -> M=v+8; N = l%16
// ---------------------------------------------------------------------------
__global__ __launch_bounds__(32)
void gemm_wmma_f32(const float* __restrict__ A,
                   const float* __restrict__ Bm,
                   const float* __restrict__ bias,
                   float* __restrict__ C,
                   int M, int N, int K, int leaky)
{
    const int lane = threadIdx.x;       // 0..31
    const int half = lane >> 4;         // 0 | 1
    const int l16  = lane & 15;

    // Shifted edge tiles: always fully in-bounds.
    int n0 = blockIdx.x * 64; if (n0 > N - 64) n0 = N - 64;
    int m0 = blockIdx.y * 16; if (m0 > M - 16) m0 = M - 16;

    // Accumulators preloaded with per-column bias (broadcast over rows).
    v8f acc[4];
    #pragma unroll
    for (int t = 0; t < 4; ++t) {
        const float bv = bias[n0 + 16 * t + l16];
        #pragma unroll
        for (int v = 0; v < 8; ++v) acc[t][v] = bv;
    }

    // Per-lane base pointers; K-loop uses immediate offsets / pointer bumps.
    const float* Ap  = A + (size_t)(m0 + l16) * K + 2 * half;   // 8B aligned (K even)
    const float* Bp0 = Bm + (size_t)(2 * half) * N + n0 + l16;  // row k0 + 2*half
    const float* Bp1 = Bp0 + N;                                 // row k0 + 2*half + 1
    const size_t stepB = (size_t)4 * N;

    #pragma unroll 2
    for (int k0 = 0; k0 < K; k0 += 4) {
        const v2f a = *(const v2f*)(Ap + k0);   // global_load_b64

        v2f b[4];
        #pragma unroll
        for (int t = 0; t < 4; ++t) {           // immediate offsets 0/64/128/192B
            b[t].x = Bp0[16 * t];
            b[t].y = Bp1[16 * t];
        }
        Bp0 += stepB;
        Bp1 += stepB;

        #pragma unroll
        for (int t = 0; t < 4; ++t)
            acc[t] = __builtin_amdgcn_wmma_f32_16x16x4_f32(
                /*neg_a=*/false, a, /*neg_b=*/false, b[t],
                /*c_mod=*/(short)0, acc[t], /*reuse_a=*/false, /*reuse_b=*/false);
    }

    // Unconditional stores (tile fully in-bounds by construction).
    #pragma unroll
    for (int t = 0; t < 4; ++t) {
        float* Cp = C + (size_t)(m0 + 8 * half) * N + (n0 + 16 * t + l16);
        #pragma unroll
        for (int v = 0; v < 8; ++v) {           // row = m0 + 8*half + v
            float x = acc[t][v];
            if (leaky) x = (x >= 0.0f) ? x : x * LRELU_SLOPE;
            Cp[(size_t)v * N] = x;
        }
    }
}

// ---------------------------------------------------------------------------
// Per-(b,g) sine-MLP decode. One block per b; the 8450-float generated
// weight row is staged once into LDS (33.8 KB of the 320 KB WGP LDS) and
// shared by the G=100 threads. Each (i,j) MAC reads a single LDS word
// uniformly across the wave -> broadcast, no bank conflicts.
// wab row layout: w0[64] | b0[1] | w1[64*64] | b1[64] | w2[64*64] | b2[64]
//                 | w3[64] | b3[1]
// ---------------------------------------------------------------------------
__global__ __launch_bounds__(128)
void decode_kernel(const float* __restrict__ logP,
                   const float* __restrict__ wab,
                   float* __restrict__ out, int G)
{
    __shared__ float sW[OUT_PER_B];
    const int b   = blockIdx.x;
    const int tid = threadIdx.x;

    const float* Wrow = wab + (size_t)b * OUT_PER_B;
    for (int i = tid; i < OUT_PER_B; i += blockDim.x) sW[i] = Wrow[i];
    __syncthreads();

    if (tid >= G) return;

    const float* w0 = sW;                 // 64
    const float  b0 = sW[64];
    const float* w1 = sW + 65;            // 64x64 row-major (di,do)
    const float* b1 = sW + 65 + 4096;     // 64
    const float* w2 = sW + 4225;          // 64x64
    const float* b2 = sW + 8321;          // 64
    const float* w3 = sW + 8385;          // 64
    const float  b3 = sW[8449];

    // x = (log_P - LOGP_MEAN) / LOGP_STD,  LOGP_MEAN=0, LOGP_STD=3
    const float x0 = logP[b * G + tid] * (1.0f / 3.0f);

    float ha[64], hb[64];

    // layer 0: (1x64) weight, scalar bias, *30, sin
    #pragma unroll 8
    for (int j = 0; j < 64; ++j)
        ha[j] = sinf(30.0f * (x0 * w0[j] + b0));

    // layer 1
    for (int j = 0; j < 64; ++j) {
        float acc = b1[j];
        #pragma unroll 8
        for (int i = 0; i < 64; ++i) acc += ha[i] * w1[i * 64 + j];
        hb[j] = sinf(acc);
    }

    // layer 2
    for (int j = 0; j < 64; ++j) {
        float acc = b2[j];
        #pragma unroll 8
        for (int i = 0; i < 64; ++i) acc += hb[i] * w2[i * 64 + j];
        ha[j] = sinf(acc);
    }

    // layer 3: (64x1) weight, scalar bias
    float acc = b3;
    #pragma unroll 8
    for (int i = 0; i < 64; ++i) acc += ha[i] * w3[i];

    // T_pred * T_STD + T_MEAN
    out[b * G + tid] = acc * 500.0f + 1500.0f;
}

// ---------------------------------------------------------------------------
// Launch: 4 WMMA GEMMs (trunk computed per-b only) + decode.
// Workspace: bufA(128x512) | bufB(128x512) | wab(128x8450)  ~= 4.85 MB
// ---------------------------------------------------------------------------
extern "C" void kernel_launch(void* const* d_in, const int* in_sizes, int n_in,
                              void* d_out, int out_size, void* d_ws, size_t ws_size,
                              hipStream_t stream)
{
    const float* z   = (const float*)d_in[0];  // (B, 16)
    const float* lgP = (const float*)d_in[1];  // (B, G)
    const float* hW0 = (const float*)d_in[2];  // (16, 512)
    const float* hb0 = (const float*)d_in[3];  // (512)
    const float* hW1 = (const float*)d_in[4];  // (512, 512)
    const float* hb1 = (const float*)d_in[5];  // (512)
    const float* hW2 = (const float*)d_in[6];  // (512, 512)
    const float* hb2 = (const float*)d_in[7];  // (512)
    const float* hWo = (const float*)d_in[8];  // (512, 8450)
    const float* hbo = (const float*)d_in[9];  // (8450)
    float* out = (float*)d_out;

    const int L   = 16;
    const int HL  = 512;
    const int Bsz = in_sizes[0] / L;           // 128
    const int G   = in_sizes[1] / Bsz;         // 100
    const int OUTS = OUT_PER_B;                // 8450

    float* bufA = (float*)d_ws;
    float* bufB = bufA + (size_t)Bsz * HL;
    float* wab  = bufB + (size_t)Bsz * HL;

    auto tilesN = [](int n) { return (n + 63) / 64; };
    auto tilesM = [](int m) { return (m + 15) / 16; };
    const dim3 blk(32);

    // h0 = lrelu(z @ hW0 + hb0)          (128 x 512, K=16)
    gemm_wmma_f32<<<dim3(tilesN(HL), tilesM(Bsz)), blk, 0, stream>>>(
        z, hW0, hb0, bufA, Bsz, HL, L, 1);
    // h1 = lrelu(h0 @ hW1 + hb1)         (128 x 512, K=512)
    gemm_wmma_f32<<<dim3(tilesN(HL), tilesM(Bsz)), blk, 0, stream>>>(
        bufA, hW1, hb1, bufB, Bsz, HL, HL, 1);
    // h2 = lrelu(h1 @ hW2 + hb2)         (128 x 512, K=512)
    gemm_wmma_f32<<<dim3(tilesN(HL), tilesM(Bsz)), blk, 0, stream>>>(
        bufB, hW2, hb2, bufA, Bsz, HL, HL, 1);
    // wab = h2 @ hWo + hbo               (128 x 8450, K=512)
    gemm_wmma_f32<<<dim3(tilesN(OUTS), tilesM(Bsz)), blk, 0, stream>>>(
        bufA, hWo, hbo, wab, Bsz, OUTS, HL, 0);

    // per-(b,g) generated sine-MLP
    decode_kernel<<<dim3(Bsz), dim3(128), 0, stream>>>(lgP, wab, out, G);
}